// SparseGraphTransformerLayer_17248588660969
// MI455X (gfx1250) — compile-verified
//
#include <hip/hip_runtime.h>
#include <hip/hip_bf16.h>
#include <math.h>

// SparseGraphTransformerLayer for MI455X (gfx1250, wave32).
// fp32 end-to-end with V_WMMA_F32_16X16X4_F32 (CDNA5's fp32 WMMA), weights
// pre-transposed so every matrix fragment load is a contiguous b64.

#define D_FEAT 256
#define LN_EPS 1e-5f

typedef float v2f __attribute__((ext_vector_type(2)));
typedef float v8f __attribute__((ext_vector_type(8)));

__device__ __forceinline__ float gelu_tanh(float x) {
  // jax.nn.gelu default (approximate=True)
  float x3 = x * x * x;
  return 0.5f * x * (1.0f + tanhf(0.7978845608028654f * (x + 0.044715f * x3)));
}

// Monotonic float <-> uint mapping so u32 atomicMax implements segment-max.
__device__ __forceinline__ unsigned f2ord(float f) {
  unsigned u = __float_as_uint(f);
  return (u & 0x80000000u) ? ~u : (u | 0x80000000u);
}
__device__ __forceinline__ float ord2f(unsigned u) {
  return (u & 0x80000000u) ? __uint_as_float(u & 0x7FFFFFFFu)
                           : __uint_as_float(~u);
}

// ---------------------------------------------------------------------------
// LDS-tiled transpose: in [K][Nout] -> out [Nout][K]. grid (Nout/16, K/16),
// 256 threads = one 16x16 tile, coalesced on both sides.
// ---------------------------------------------------------------------------
__global__ __launch_bounds__(256) void transpose_kernel(
    const float* __restrict__ in, float* __restrict__ out, int K, int Nout) {
  __shared__ float tile[16][17];
  const int tx = threadIdx.x & 15, ty = threadIdx.x >> 4;
  const int n0 = blockIdx.x << 4, k0 = blockIdx.y << 4;
  tile[ty][tx] = in[(size_t)(k0 + ty) * Nout + (n0 + tx)];
  __syncthreads();
  out[(size_t)(n0 + ty) * K + (k0 + tx)] = tile[tx][ty];
}

// ---------------------------------------------------------------------------
// GEMM with pre-transposed weights:
//   C = act(A[M,K] @ W[K,*] + bias),  Wt = W^T laid out [NCOLS][K] row-major.
// One wave32 computes a 16x32 output tile (two WMMA accumulators sharing the
// A fragment). 8 waves/block -> 256 global columns per block.
// Global column gcol maps to output matrix gcol/OUT_W (stride mat_stride
// floats) at column gcol%OUT_W — this lets one launch produce Q,K,V,S from a
// packed [4*D][K] weight block.
// Fragments (ISA 7.12.2, f32):
//   A 16x4 : lane l -> row l&15, K = 2*(l>>4) (+1)   -> contiguous b64
//   B 4x16 : lane l -> col l&15, K = 2*(l>>4) (+1)   -> contiguous b64 in Wt
//   C 16x16: vgpr r -> row r + 8*(l>>4), col l&15
// ---------------------------------------------------------------------------
template <int K, int OUT_W, bool GELU>
__global__ __launch_bounds__(256) void gemm_wmma_f32(
    const float* __restrict__ A, const float* __restrict__ Wt,
    const float* __restrict__ biasPack, float* __restrict__ C,
    size_t mat_stride) {
  const int wave  = threadIdx.x >> 5;
  const int lane  = threadIdx.x & 31;
  const int row0  = blockIdx.x << 4;
  const int gcol0 = ((blockIdx.y << 3) + wave) << 5;  // 32 cols per wave
  const int mn    = lane & 15;
  const int koff  = (lane >> 4) << 1;  // 0 or 2

  const float* arow  = A + (size_t)(row0 + mn) * K + koff;
  const float* brow0 = Wt + (size_t)(gcol0 + mn) * K + koff;
  const float* brow1 = brow0 + (size_t)16 * K;

  v8f accs[2] = {{}, {}};
#pragma unroll 8
  for (int k0 = 0; k0 < K; k0 += 4) {
    v2f a  = *(const v2f*)(arow + k0);
    v2f b0 = *(const v2f*)(brow0 + k0);
    v2f b1 = *(const v2f*)(brow1 + k0);
    accs[0] = __builtin_amdgcn_wmma_f32_16x16x4_f32(
        false, a, false, b0, (short)0, accs[0], false, false);
    accs[1] = __builtin_amdgcn_wmma_f32_16x16x4_f32(
        false, a, false, b1, (short)0, accs[1], false, false);
  }

  const int rbase = row0 + ((lane >> 4) << 3);
#pragma unroll
  for (int t = 0; t < 2; ++t) {
    const int gcol = gcol0 + (t << 4) + mn;
    float* Cm      = C + (size_t)(gcol / OUT_W) * mat_stride;
    const int col  = gcol % OUT_W;  // OUT_W is a power of two
    const float bn = biasPack[gcol];
#pragma unroll
    for (int r = 0; r < 8; ++r) {
      float val = accs[t][r] + bn;
      if (GELU) val = gelu_tanh(val);
      Cm[(size_t)(rbase + r) * OUT_W + col] = val;
    }
  }
}

// ---------------------------------------------------------------------------
// Edge phase: one wave32 per edge; lanes stride the 256-wide feature dim.
// ---------------------------------------------------------------------------
__global__ __launch_bounds__(256) void edge_scores_kernel(
    const float* __restrict__ q, const float* __restrict__ k,
    const int* __restrict__ ei, const float* __restrict__ ew,
    const float* __restrict__ we, const float* __restrict__ be,
    float* __restrict__ scores, unsigned* __restrict__ segmax, int E) {
  const int e    = (int)((blockIdx.x * blockDim.x + threadIdx.x) >> 5);
  const int lane = threadIdx.x & 31;
  if (e >= E) return;
  const int src = ei[e];      // edge_index[0]
  const int dst = ei[E + e];  // edge_index[1]
  const float4* qv = (const float4*)(q + (size_t)dst * D_FEAT) + lane * 2;
  const float4* kv = (const float4*)(k + (size_t)src * D_FEAT) + lane * 2;
  float4 a0 = qv[0], a1 = qv[1];
  float4 b0 = kv[0], b1 = kv[1];
  float acc = a0.x * b0.x + a0.y * b0.y + a0.z * b0.z + a0.w * b0.w +
              a1.x * b1.x + a1.y * b1.y + a1.z * b1.z + a1.w * b1.w;
#pragma unroll
  for (int o = 16; o; o >>= 1) acc += __shfl_xor(acc, o, 32);
  if (lane == 0) {
    float sc = acc * 0.0625f /* D^-0.5 */ + ew[e] * we[0] + be[0];
    scores[e] = sc;
    atomicMax(&segmax[dst], f2ord(sc));
  }
}

__global__ __launch_bounds__(256) void edge_exp_kernel(
    const int* __restrict__ ei, const float* __restrict__ scores,
    const unsigned* __restrict__ segmax, float* __restrict__ exps,
    float* __restrict__ denom, int E) {
  const int e = (int)(blockIdx.x * blockDim.x + threadIdx.x);
  if (e >= E) return;
  const int dst = ei[E + e];
  float ex = expf(scores[e] - ord2f(segmax[dst]));
  exps[e] = ex;
  atomicAdd(&denom[dst], ex);
}

__global__ __launch_bounds__(256) void edge_scatter_kernel(
    const float* __restrict__ v, const int* __restrict__ ei,
    const float* __restrict__ exps, const float* __restrict__ denom,
    float* __restrict__ outacc, int E) {
  const int e    = (int)((blockIdx.x * blockDim.x + threadIdx.x) >> 5);
  const int lane = threadIdx.x & 31;
  if (e >= E) return;
  const int src  = ei[e];
  const int dst  = ei[E + e];
  const float at = exps[e] / fmaxf(denom[dst], 1e-12f);
  const float4* vv = (const float4*)(v + (size_t)src * D_FEAT) + lane * 2;
  float4 m0 = vv[0], m1 = vv[1];
  float* o = outacc + (size_t)dst * D_FEAT + lane * 8;
  atomicAdd(o + 0, m0.x * at);
  atomicAdd(o + 1, m0.y * at);
  atomicAdd(o + 2, m0.z * at);
  atomicAdd(o + 3, m0.w * at);
  atomicAdd(o + 4, m1.x * at);
  atomicAdd(o + 5, m1.y * at);
  atomicAdd(o + 6, m1.z * at);
  atomicAdd(o + 7, m1.w * at);
}

// ---------------------------------------------------------------------------
// Residual add + LayerNorm, one 256-thread block (8 wave32) per row.
// ---------------------------------------------------------------------------
__global__ __launch_bounds__(256) void residual_ln_kernel(
    const float* __restrict__ a, const float* __restrict__ b,
    const float* __restrict__ c, const float* __restrict__ g,
    const float* __restrict__ beta, float* __restrict__ out) {
  const int row = blockIdx.x, tid = threadIdx.x;
  const size_t idx = (size_t)row * D_FEAT + tid;
  float r = a[idx] + b[idx];
  if (c) r += c[idx];
  float sum = r, sq = r * r;
#pragma unroll
  for (int o = 16; o; o >>= 1) {
    sum += __shfl_xor(sum, o, 32);
    sq  += __shfl_xor(sq, o, 32);
  }
  __shared__ float ssum[8], ssq[8];
  const int w = tid >> 5, l = tid & 31;
  if (l == 0) { ssum[w] = sum; ssq[w] = sq; }
  __syncthreads();
  if (tid < 32) {
    float ts = (l < 8) ? ssum[l] : 0.0f;
    float tq = (l < 8) ? ssq[l] : 0.0f;
#pragma unroll
    for (int o = 4; o; o >>= 1) {
      ts += __shfl_xor(ts, o, 32);
      tq += __shfl_xor(tq, o, 32);
    }
    if (l == 0) { ssum[0] = ts; ssq[0] = tq; }
  }
  __syncthreads();
  const float mu   = ssum[0] * (1.0f / D_FEAT);
  const float var  = ssq[0] * (1.0f / D_FEAT) - mu * mu;
  const float rstd = rsqrtf(var + LN_EPS);
  out[idx] = (r - mu) * rstd * g[tid] + beta[tid];
}

// ---------------------------------------------------------------------------
extern "C" void kernel_launch(void* const* d_in, const int* in_sizes, int n_in,
                              void* d_out, int out_size, void* d_ws,
                              size_t ws_size, hipStream_t stream) {
  const float* x    = (const float*)d_in[0];
  const int*   ei   = (const int*)d_in[1];   // edge_index flat [2,E]
  const float* ew   = (const float*)d_in[2];
  const float* wq   = (const float*)d_in[3];
  const float* bq   = (const float*)d_in[4];
  const float* wk   = (const float*)d_in[5];
  const float* bk   = (const float*)d_in[6];
  const float* wv   = (const float*)d_in[7];
  const float* bv   = (const float*)d_in[8];
  const float* wsm  = (const float*)d_in[9];
  const float* bs   = (const float*)d_in[10];
  const float* we   = (const float*)d_in[11];
  const float* be   = (const float*)d_in[12];
  const float* g1   = (const float*)d_in[13];
  const float* b1   = (const float*)d_in[14];
  const float* g2   = (const float*)d_in[15];
  const float* b2   = (const float*)d_in[16];
  const float* wff1 = (const float*)d_in[17];
  const float* bff1 = (const float*)d_in[18];
  const float* wff2 = (const float*)d_in[19];
  const float* bff2 = (const float*)d_in[20];

  const int D = D_FEAT;
  const int N = in_sizes[0] / D;  // 10000 (multiple of 16)
  const int E = in_sizes[2];      // 320000
  const size_t ND = (size_t)N * D;
  const size_t DD = (size_t)D * D;

  // Workspace: q|k|v|s|x1|attn | scores|exps|denom|segmax | wt's | biasPack
  float* q      = (float*)d_ws;
  float* kbuf   = q + ND;
  float* vbuf   = kbuf + ND;
  float* sbuf   = vbuf + ND;
  float* x1     = sbuf + ND;
  float* attn   = x1 + ND;
  float* scores = attn + ND;
  float* exps   = scores + E;
  float* denom  = exps + E;
  unsigned* segmax = (unsigned*)(denom + N);
  float* wtqkvs = (float*)(segmax + N);  // packed [4D][D] = 4 transposed W's
  float* wt1    = wtqkvs + 4 * DD;       // [2D][D]
  float* wt2    = wt1 + 2 * DD;          // [D][2D]
  float* bpack  = wt2 + 2 * DD;          // [4D] packed q/k/v/s biases
  float* h      = q;     // FFN hidden [N,2D] reuses q+kbuf after edge phase
  float* y2     = vbuf;  // FFN output [N,D]  reuses v after scatter

  hipMemsetAsync(attn, 0, ND * sizeof(float), stream);
  hipMemsetAsync(denom, 0, N * sizeof(float), stream);
  hipMemsetAsync(segmax, 0, N * sizeof(unsigned), stream);  // key 0 == -inf

  dim3 blk(256);

  // --- one-time (cheap) weight transposes + bias packing --------------------
  transpose_kernel<<<dim3(16, 16), blk, 0, stream>>>(wq,  wtqkvs + 0 * DD, D, D);
  transpose_kernel<<<dim3(16, 16), blk, 0, stream>>>(wk,  wtqkvs + 1 * DD, D, D);
  transpose_kernel<<<dim3(16, 16), blk, 0, stream>>>(wv,  wtqkvs + 2 * DD, D, D);
  transpose_kernel<<<dim3(16, 16), blk, 0, stream>>>(wsm, wtqkvs + 3 * DD, D, D);
  transpose_kernel<<<dim3(32, 16), blk, 0, stream>>>(wff1, wt1, D, 2 * D);
  transpose_kernel<<<dim3(16, 32), blk, 0, stream>>>(wff2, wt2, 2 * D, D);
  hipMemcpyAsync(bpack + 0 * D, bq, D * sizeof(float), hipMemcpyDeviceToDevice, stream);
  hipMemcpyAsync(bpack + 1 * D, bk, D * sizeof(float), hipMemcpyDeviceToDevice, stream);
  hipMemcpyAsync(bpack + 2 * D, bv, D * sizeof(float), hipMemcpyDeviceToDevice, stream);
  hipMemcpyAsync(bpack + 3 * D, bs, D * sizeof(float), hipMemcpyDeviceToDevice, stream);

  const int rowTiles = N / 16;  // 625

  // --- fused Q/K/V/skip: one pass over x, 1024 packed output columns --------
  gemm_wmma_f32<256, 256, false>
      <<<dim3(rowTiles, 4), blk, 0, stream>>>(x, wtqkvs, bpack, q, ND);

  // --- per-edge scores + segment softmax + scatter-add ----------------------
  edge_scores_kernel<<<(E + 7) / 8, blk, 0, stream>>>(q, kbuf, ei, ew, we, be,
                                                      scores, segmax, E);
  edge_exp_kernel<<<(E + 255) / 256, blk, 0, stream>>>(ei, scores, segmax,
                                                       exps, denom, E);
  edge_scatter_kernel<<<(E + 7) / 8, blk, 0, stream>>>(vbuf, ei, exps, denom,
                                                       attn, E);

  // --- x1 = LN(x + attn + skip) ---------------------------------------------
  residual_ln_kernel<<<N, blk, 0, stream>>>(x, attn, sbuf, g1, b1, x1);

  // --- FFN: h = gelu(x1 @ wff1 + bff1); y2 = h @ wff2 + bff2 ----------------
  gemm_wmma_f32<256, 512, true>
      <<<dim3(rowTiles, 2), blk, 0, stream>>>(x1, wt1, bff1, h, 0);
  gemm_wmma_f32<512, 256, false>
      <<<dim3(rowTiles, 1), blk, 0, stream>>>(h, wt2, bff2, y2, 0);

  // --- out = LN(x1 + y2) ----------------------------------------------------
  residual_ln_kernel<<<N, blk, 0, stream>>>(x1, y2, nullptr, g2, b2,
                                            (float*)d_out);
}